// BuildModel_82325933129924
// MI455X (gfx1250) — compile-verified
//
#include <hip/hip_runtime.h>
#include <hip/hip_bf16.h>
#include <math.h>

// ---- problem constants (from reference) ----
#define BB   8
#define DD   59
#define CC   80
#define FH_  16
#define FW_  44
#define NPIX (BB*FH_*FW_)          // 5632 pixels, one wave32 each
#define NX_  200
#define NY_  200
#define NCH_ (DD + CC)             // 139
#define CH_STRIDE (FH_*FW_)        // 704 floats between channels
#define OUT_PER_B (NY_*CC*NX_)     // 3,200,000 floats per batch

typedef __attribute__((ext_vector_type(2))) float v2f;
typedef __attribute__((ext_vector_type(8))) float v8f;

__global__ void zero_out_kernel(float4* __restrict__ out, int n4) {
  int i = blockIdx.x * blockDim.x + threadIdx.x;
  if (i < n4) out[i] = make_float4(0.f, 0.f, 0.f, 0.f);
}

__global__ __launch_bounds__(256) void bevpool_wmma_kernel(
    const float* __restrict__ dno,    // (B, 139, 16, 44)
    const float* __restrict__ intr,   // (B, 3, 3)
    float* __restrict__ out)          // (B, NY*C, 1, NX) flat
{
  const int lane  = threadIdx.x & 31;
  const int wid   = blockIdx.x * (blockDim.x >> 5) + (threadIdx.x >> 5); // pixel id, exact fit
  const int b     = wid / (FH_ * FW_);
  const int hw    = wid - b * (FH_ * FW_);
  const int h     = hw / FW_;
  const int w     = hw - h * FW_;

  // ---- Kinv (3x3 adjugate inverse), then G = Kinv * (px, py, 1) ----
  const float* K = intr + b * 9;
  float a0 = K[0], a1 = K[1], a2 = K[2];
  float a3 = K[3], a4 = K[4], a5 = K[5];
  float a6 = K[6], a7 = K[7], a8 = K[8];
  float c00 =  (a4 * a8 - a5 * a7);
  float c10 = -(a3 * a8 - a5 * a6);
  float c20 =  (a3 * a7 - a4 * a6);
  float det = a0 * c00 + a1 * c10 + a2 * c20;
  float inv = 1.0f / det;
  float k00 = c00 * inv;
  float k01 = -(a1 * a8 - a2 * a7) * inv;
  float k02 =  (a1 * a5 - a2 * a4) * inv;
  float k10 = c10 * inv;
  float k11 =  (a0 * a8 - a2 * a6) * inv;
  float k12 = -(a0 * a5 - a2 * a3) * inv;
  float k20 = c20 * inv;
  float k21 = -(a0 * a7 - a1 * a6) * inv;
  float k22 =  (a0 * a4 - a1 * a3) * inv;

  const float px = (float)w * (703.0f / 43.0f);  // linspace(0,703,44)
  const float py = (float)h * 17.0f;             // linspace(0,255,16)
  // geom(d) = ds * G ; fold voxelization affine into pre-scaled constants:
  //   ix = trunc((ds*G0 + 50)*2) = trunc(ds*(2*G0) + 100)
  const float G0s = 2.0f * (k00 * px + k01 * py + k02);
  const float G1s = 2.0f * (k10 * px + k11 * py + k12);
  const float G2s = (k20 * px + k21 * py + k22) * (1.0f / 64.0f);

  // ---- softmax over D=59 depth logits; lane holds d=lane and d=lane+32 ----
  const float* dp = dno + ((size_t)b * NCH_) * CH_STRIDE + h * FW_ + w;
  float v0 = dp[lane * CH_STRIDE];
  bool  hi_valid = (lane + 32) < DD;
  float v1 = hi_valid ? dp[(lane + 32) * CH_STRIDE] : -INFINITY;
  float m = fmaxf(v0, v1);
  #pragma unroll
  for (int o = 16; o >= 1; o >>= 1) m = fmaxf(m, __shfl_xor(m, o, 32));
  float e0 = __expf(v0 - m);
  float e1 = hi_valid ? __expf(v1 - m) : 0.0f;
  float s = e0 + e1;
  #pragma unroll
  for (int o = 16; o >= 1; o >>= 1) s += __shfl_xor(s, o, 32);
  float rs = 1.0f / s;
  float p0 = e0 * rs;                 // depth prob for d = lane
  float p1 = e1 * rs;                 // depth prob for d = lane+32

  // ---- feat channels; lane holds c=lane, c=lane+32, (lane<16: c=lane+64) ----
  const float* fp = dno + ((size_t)(b * NCH_ + DD)) * CH_STRIDE + h * FW_ + w;
  float f0 = fp[lane * CH_STRIDE];
  float f1 = fp[(lane + 32) * CH_STRIDE];
  float f2 = (lane < 16) ? fp[(lane + 64) * CH_STRIDE] : 0.0f;

  const int  N      = lane & 15;
  const bool lolane = (lane < 16);
  const int  mbase  = lolane ? 0 : 8;

  // WMMA B operands: feat[16*ct + N] on lanes 0-15, K=0 slice only
  float bsel[5];
  bsel[0] = __shfl(f0, N, 32);
  bsel[1] = __shfl(f0, N + 16, 32);
  bsel[2] = __shfl(f1, N, 32);
  bsel[3] = __shfl(f1, N + 16, 32);
  bsel[4] = __shfl(f2, N, 32);
  // WMMA A operands: depth[16*t + M] on lanes 0-15 (M = lane), K=0 slice only
  float asel[4];
  asel[0] = __shfl(p0, N, 32);
  asel[1] = __shfl(p0, N + 16, 32);
  asel[2] = __shfl(p1, N, 32);
  asel[3] = __shfl(p1, N + 16, 32);

  const v8f zacc = {0.f, 0.f, 0.f, 0.f, 0.f, 0.f, 0.f, 0.f};
  const unsigned outb = (unsigned)(b * OUT_PER_B);

  #pragma unroll
  for (int t = 0; t < 4; ++t) {
    v2f A;
    A[0] = lolane ? asel[t] : 0.0f;   // K=0 (lanes 0-15); K=2 lanes zeroed
    A[1] = 0.0f;                      // K=1 / K=3 zero

    // voxel destination offsets for this lane's 8 D-rows (d = 16t + r + mbase)
    // kept fraction is ~100%; dropped rows scatter +0.0f to out[coff] instead
    // of branching, so EXEC stays all-1s through the whole kernel.
    // unsigned 32-bit offsets (max 25.6M) -> SADDR-form global atomics.
    unsigned voff[8];
    bool     vk[8];
    #pragma unroll
    for (int r = 0; r < 8; ++r) {
      int d = 16 * t + r + mbase;
      float dsv = 4.0f + (float)d;    // arange(4,63)
      int ix = (int)fmaf(dsv, G0s, 100.0f);   // truncation matches astype(int32)
      int iy = (int)fmaf(dsv, G1s, 100.0f);
      int iz = (int)(dsv * G2s);
      bool kept = (d < DD) & ((unsigned)ix < (unsigned)NX_) &
                  ((unsigned)iy < (unsigned)NY_) & (iz == 0);
      vk[r]   = kept;
      voff[r] = kept ? (outb + (unsigned)(iy * (CC * NX_) + ix)) : 0u;
    }

    #pragma unroll
    for (int ct = 0; ct < 5; ++ct) {
      v2f Bv;
      Bv[0] = lolane ? bsel[ct] : 0.0f;
      Bv[1] = 0.0f;
      // D[M,N] = depth[16t+M] * feat[16ct+N]   (only K=0 contributes)
      v8f dres = __builtin_amdgcn_wmma_f32_16x16x4_f32(
          /*neg_a=*/false, A, /*neg_b=*/false, Bv,
          /*c_mod=*/(short)0, zacc, /*reuse_a=*/false, /*reuse_b=*/false);
      unsigned coff = (unsigned)((16 * ct + N) * NX_);
      #pragma unroll
      for (int r = 0; r < 8; ++r) {
        float val = vk[r] ? dres[r] : 0.0f;     // predicate the value, not the op
        atomicAdd(out + (voff[r] + coff), val);
      }
    }
  }
}

extern "C" void kernel_launch(void* const* d_in, const int* in_sizes, int n_in,
                              void* d_out, int out_size, void* d_ws, size_t ws_size,
                              hipStream_t stream) {
  const float* dno  = (const float*)d_in[0];   // depthnet_out (8,139,16,44) f32
  const float* intr = (const float*)d_in[1];   // intrins (8,3,3) f32
  float* out = (float*)d_out;                  // 25,600,000 f32

  // 1) zero the accumulation grid with 16B stores (25,600,000 % 4 == 0)
  int n4 = out_size >> 2;
  zero_out_kernel<<<(n4 + 255) / 256, 256, 0, stream>>>((float4*)out, n4);

  // 2) one wave32 per pixel: softmax + WMMA outer product + atomic scatter
  //    5632 pixels / 8 waves per 256-thread block = 704 blocks (exact)
  bevpool_wmma_kernel<<<NPIX / 8, 256, 0, stream>>>(dno, intr, out);
}